// BevPoolV2Module_44032004718768
// MI455X (gfx1250) — compile-verified
//
#include <hip/hip_runtime.h>

typedef unsigned int u32x4 __attribute__((ext_vector_type(4)));
typedef int          i32x4 __attribute__((ext_vector_type(4)));
typedef int          i32x8 __attribute__((ext_vector_type(8)));

// BevPoolV2 constants (match the reference exactly)
#define NN    6
#define DDim  118
#define HH    32
#define WWd   88
#define CC    128
#define OH_   256
#define OW_   256
#define HW_   (HH * WWd)          // 2816
#define DHW_  (DDim * HW_)        // 332288
#define OHW_  (OH_ * OW_)         // 65536
#define NROWS (NN * HW_)          // 16896 feat rows (fits 16-bit TDM gather index)
#define WAVES_PER_BLOCK 8
#define CHUNK 8                   // rows per TDM gather descriptor (<=16 allowed)

// ---------------------------------------------------------------------------
// bev_pool: each wave32 owns TWO consecutive intervals with two LDS buffers.
// TDM gather-mode tensor_load_to_lds pulls each interval's feat rows (512 B
// apiece) into LDS in one DMA; in-order TENSORcnt lets us consume buffer 0
// under s_wait_tensorcnt 1 while buffer 1's DMA is still in flight.
// Lane l owns channels 4l..4l+3; scatter via no-return f32 HW atomics.
// ---------------------------------------------------------------------------
__global__ __launch_bounds__(256) void bev_pool(const float* __restrict__ feat,
                                                const float* __restrict__ depth,
                                                const int*   __restrict__ indices,
                                                const int*   __restrict__ intervals,
                                                float*       __restrict__ out,
                                                int M) {
    // 8 waves * 2 buffers * 8 rows * 128 f32 = 64 KB LDS (of 320 KB/WGP)
    __shared__ float ldsbuf[WAVES_PER_BLOCK * 2 * CHUNK * CC];

    const unsigned lane = threadIdx.x & 31u;
    const unsigned wave = threadIdx.x >> 5;
    const int seg0 = (int)((blockIdx.x * WAVES_PER_BLOCK + wave) * 2);
    if (seg0 >= M) return;                                  // wave-uniform exit
    const bool has1 = (seg0 + 1) < M;

    float*   lds0    = &ldsbuf[(wave * 2 + 0) * CHUNK * CC];
    float*   lds1    = &ldsbuf[(wave * 2 + 1) * CHUNK * CC];
    unsigned ldsoff0 = (unsigned)(size_t)lds0;              // WG LDS byte offsets
    unsigned ldsoff1 = (unsigned)(size_t)lds1;
    const unsigned long long fb = (unsigned long long)(size_t)feat;

    // Build + issue one gather-mode D# for rows of points [p, p+cnt).
    auto issue_chunk = [&](int p, int cnt, unsigned ldsoff, float* dv) {
        unsigned rows[CHUNK];
        unsigned last = 0;
#pragma unroll
        for (int j = 0; j < CHUNK; ++j) {
            if (j < cnt) {
                unsigned idx = (unsigned)__builtin_amdgcn_readfirstlane(indices[p + j]);
                rows[j] = (idx / (unsigned)DHW_) * (unsigned)HW_ + (idx % (unsigned)HW_);
                dv[j]   = depth[idx];                        // uniform -> SMEM load
                last    = rows[j];
            } else {                                         // pad (ignored: tile_dim1=cnt)
                rows[j] = last;
                dv[j]   = 0.f;
            }
        }
        u32x4 g0;
        g0.x = 0x80000001u;                                  // count=1 | gather_mode, 16-bit idx
        g0.y = ldsoff;                                       // lds_addr
        g0.z = (unsigned)(fb & 0xFFFFFFFFull);               // global_addr[31:0]
        g0.w = (unsigned)((fb >> 32) & 0x1FFFFFFull)         // global_addr[56:32]
             | 0x80000000u;                                  // type=2 ("image")
        i32x8 g1;
        g1[0] = 0x20000;                                     // data_size=2 (4 B), wg_mask=0
        g1[1] = (int)((unsigned)CC << 16);                   // tensor_dim0 = 128
        g1[2] = (int)((unsigned)NROWS << 16);                // tensor_dim1 = 16896 rows
        g1[3] = (int)((unsigned)CC << 16);                   // tile_dim0 = 128
        g1[4] = cnt;                                         // tile_dim1 = #valid indices
        g1[5] = CC;                                          // tensor_dim0_stride = 128
        g1[6] = 0;
        g1[7] = 0;
        i32x4 g2, g3;
        g2[0] = (int)(rows[0] | (rows[1] << 16));            // row_index_0..7 (16-bit)
        g2[1] = (int)(rows[2] | (rows[3] << 16));
        g2[2] = (int)(rows[4] | (rows[5] << 16));
        g2[3] = (int)(rows[6] | (rows[7] << 16));
        const int padw = (int)(last | (last << 16));         // row_index_8..15 (unused)
        g3[0] = padw; g3[1] = padw; g3[2] = padw; g3[3] = padw;
#if __has_builtin(__builtin_amdgcn_tensor_load_to_lds)
  #if defined(__clang_major__) && (__clang_major__ >= 23)
        __builtin_amdgcn_tensor_load_to_lds(g0, g1, g2, g3,
                                            (i32x8){0, 0, 0, 0, 0, 0, 0, 0}, 0);
  #else
        __builtin_amdgcn_tensor_load_to_lds(g0, g1, g2, g3, 0);
  #endif
#else
        asm volatile("tensor_load_to_lds %0, %1, %2, %3"
                     :: "s"(g0), "s"(g1), "s"(g2), "s"(g3) : "memory");
#endif
    };

    auto accum_chunk = [&](const float* lptr, const float* dv, int cnt, float4& acc) {
#pragma unroll
        for (int j = 0; j < CHUNK; ++j) {
            if (j < cnt) {
                const float4 v = *(const float4*)(lptr + j * CC + lane * 4u); // ds_load_b128
                acc.x += v.x * dv[j];
                acc.y += v.y * dv[j];
                acc.z += v.z * dv[j];
                acc.w += v.w * dv[j];
            }
        }
    };

    // Scatter into (N, C, OH, OW): channel stride is OH*OW.
    auto scatter = [&](int rank, const float4& acc) {
        const unsigned n_o = (unsigned)rank / (unsigned)OHW_;
        const unsigned pix = (unsigned)rank % (unsigned)OHW_;
        float* ob = out + (size_t)n_o * ((size_t)CC * OHW_) + pix;
        const unsigned c0 = lane * 4u;
        unsigned long long a0 = (unsigned long long)(size_t)(ob + (size_t)(c0 + 0) * OHW_);
        unsigned long long a1 = (unsigned long long)(size_t)(ob + (size_t)(c0 + 1) * OHW_);
        unsigned long long a2 = (unsigned long long)(size_t)(ob + (size_t)(c0 + 2) * OHW_);
        unsigned long long a3 = (unsigned long long)(size_t)(ob + (size_t)(c0 + 3) * OHW_);
        asm volatile("global_atomic_add_f32 %0, %1, off" :: "v"(a0), "v"(acc.x) : "memory");
        asm volatile("global_atomic_add_f32 %0, %1, off" :: "v"(a1), "v"(acc.y) : "memory");
        asm volatile("global_atomic_add_f32 %0, %1, off" :: "v"(a2), "v"(acc.z) : "memory");
        asm volatile("global_atomic_add_f32 %0, %1, off" :: "v"(a3), "v"(acc.w) : "memory");
    };

    // Segment metadata (forced to SGPRs).
    int s0 = __builtin_amdgcn_readfirstlane(intervals[3 * seg0 + 0]);
    int e0 = __builtin_amdgcn_readfirstlane(intervals[3 * seg0 + 1]);
    int r0 = __builtin_amdgcn_readfirstlane(intervals[3 * seg0 + 2]);
    int s1 = 0, e1 = 0, r1 = 0;
    if (has1) {
        s1 = __builtin_amdgcn_readfirstlane(intervals[3 * seg0 + 3]);
        e1 = __builtin_amdgcn_readfirstlane(intervals[3 * seg0 + 4]);
        r1 = __builtin_amdgcn_readfirstlane(intervals[3 * seg0 + 5]);
    }
    int c0 = e0 - s0; if (c0 > CHUNK) c0 = CHUNK; if (c0 < 0) c0 = 0;
    int c1 = has1 ? (e1 - s1) : 0; if (c1 > CHUNK) c1 = CHUNK; if (c1 < 0) c1 = 0;

    // Issue both DMAs back-to-back, then consume pipelined.
    float dv0[CHUNK], dv1[CHUNK];
    if (c0 > 0) issue_chunk(s0, c0, ldsoff0, dv0);
    if (c1 > 0) issue_chunk(s1, c1, ldsoff1, dv1);

    float4 acc0 = {0.f, 0.f, 0.f, 0.f};
    if (c0 > 0) {
        if (c1 > 0) asm volatile("s_wait_tensorcnt 0x1" ::: "memory"); // buf0 done, buf1 in flight
        else        asm volatile("s_wait_tensorcnt 0x0" ::: "memory");
        accum_chunk(lds0, dv0, c0, acc0);
        for (int p = s0 + c0; p < e0; ) {                    // overflow path (len > CHUNK)
            int cc = e0 - p; if (cc > CHUNK) cc = CHUNK;
            issue_chunk(p, cc, ldsoff0, dv0);
            asm volatile("s_wait_tensorcnt 0x0" ::: "memory");
            accum_chunk(lds0, dv0, cc, acc0);
            p += cc;
        }
    }
    scatter(r0, acc0);

    if (has1) {
        float4 acc1 = {0.f, 0.f, 0.f, 0.f};
        if (c1 > 0) {
            asm volatile("s_wait_tensorcnt 0x0" ::: "memory");
            accum_chunk(lds1, dv1, c1, acc1);
            for (int p = s1 + c1; p < e1; ) {                // overflow path
                int cc = e1 - p; if (cc > CHUNK) cc = CHUNK;
                issue_chunk(p, cc, ldsoff1, dv1);
                asm volatile("s_wait_tensorcnt 0x0" ::: "memory");
                accum_chunk(lds1, dv1, cc, acc1);
                p += cc;
            }
        }
        scatter(r1, acc1);
    }
}

// ---------------------------------------------------------------------------
// bev_zero: zero the 201 MB output (atomics accumulate; harness poisons once).
// ---------------------------------------------------------------------------
__global__ __launch_bounds__(256) void bev_zero(float4* __restrict__ out, unsigned n4) {
    unsigned i = blockIdx.x * 256u + threadIdx.x;
    if (i < n4) out[i] = float4{0.f, 0.f, 0.f, 0.f};
}

extern "C" void kernel_launch(void* const* d_in, const int* in_sizes, int n_in,
                              void* d_out, int out_size, void* d_ws, size_t ws_size,
                              hipStream_t stream) {
    (void)n_in; (void)d_ws; (void)ws_size;
    const float* feat      = (const float*)d_in[0];
    const float* depth     = (const float*)d_in[1];
    const int*   indices   = (const int*)d_in[2];
    const int*   intervals = (const int*)d_in[3];
    float*       out       = (float*)d_out;

    const int M = in_sizes[3] / 3;                  // 200000 intervals
    const unsigned n4 = (unsigned)(out_size / 4);   // 12,582,912 float4s

    bev_zero<<<(n4 + 255u) / 256u, 256, 0, stream>>>((float4*)out, n4);

    const int segs_per_block = WAVES_PER_BLOCK * 2;
    const int blocks = (M + segs_per_block - 1) / segs_per_block;
    bev_pool<<<blocks, 256, 0, stream>>>(feat, depth, indices, intervals, out, M);
}